// HierarchicalRGCN_56908316672547
// MI455X (gfx1250) — compile-verified
//
#include <hip/hip_runtime.h>

// ---------------- problem constants (match reference) ----------------
#define NN 50000
#define EE 800000
#define DD 256
#define RR 16
#define BB 8
#define LN_EPS 1e-5f
#define IDS_CAP (EE + NN + 16 * RR)   // padded bucket capacity (multiple of 16)
#define APAD 260                      // LDS A-row stride: 16B-aligned rows, conflict-free b64 reads

typedef __attribute__((ext_vector_type(2))) float v2f;
typedef __attribute__((ext_vector_type(8))) float v8f;

// fire-and-forget f32 atomic add (CDNA5 GLOBAL_ATOMIC_ADD_F32, no-return form, STOREcnt)
__device__ __forceinline__ void atomicAddF32(float* p, float v) {
  asm volatile("global_atomic_add_f32 %0, %1, off"
               :: "v"((unsigned long long)(uintptr_t)p), "v"(v) : "memory");
}

// async global -> LDS copy of 16 bytes per lane (CDNA5 GLOBAL_LOAD_ASYNC_TO_LDS_B128)
__device__ __forceinline__ void async_load16(void* lds_ptr, const void* gptr) {
  unsigned int l = (unsigned int)(uintptr_t)lds_ptr;       // low 32 bits = LDS byte offset
  unsigned long long g = (unsigned long long)(uintptr_t)gptr;
  asm volatile("global_load_async_to_lds_b128 %0, %1, off" :: "v"(l), "v"(g) : "memory");
}
__device__ __forceinline__ void async_wait0() {
  asm volatile("s_wait_asynccnt 0x0" ::: "memory");
}

// ---------------- tiny utility kernels ----------------
__global__ __launch_bounds__(256) void fill_i32(int* p, int n, int v) {
  int i = blockIdx.x * 256 + threadIdx.x;
  if (i < n) p[i] = v;
}

__global__ __launch_bounds__(256) void zero_f32(float* p, int n) {
  int i = blockIdx.x * 256 + threadIdx.x;
  if (i < n) p[i] = 0.0f;
}

// histogram of edge types
__global__ __launch_bounds__(256) void hist_kernel(const int* __restrict__ et, int* __restrict__ cntR) {
  int e = blockIdx.x * 256 + threadIdx.x;
  if (e < EE) atomicAdd(&cntR[et[e]], 1);
}

// exclusive prefix over 16 buckets, padded to multiples of 16; self-loops go to bucket R-1
__global__ void prefix_kernel(const int* __restrict__ cntR, int* __restrict__ off, int* __restrict__ cur) {
  if (threadIdx.x == 0 && blockIdx.x == 0) {
    int o = 0;
    for (int r = 0; r < RR; ++r) {
      off[r] = o;
      cur[r] = o;
      int c = cntR[r] + (r == RR - 1 ? NN : 0);
      o += ((c + 15) >> 4) << 4;
    }
    off[RR] = o;
  }
}

// scatter edge ids (real edges: id in [0,E); self-loops: id = E + n) into buckets
__global__ __launch_bounds__(256) void scatter_ids(const int* __restrict__ et, int* __restrict__ cur,
                                                   int* __restrict__ ids) {
  int i = blockIdx.x * 256 + threadIdx.x;
  if (i >= EE + NN) return;
  int t = (i < EE) ? et[i] : (RR - 1);
  int pos = atomicAdd(&cur[t], 1);
  ids[pos] = i;
}

// WT[r,o,i] = sum_b att[r,b] * basis[b,i,o]   (K-transposed so B operands are contiguous b64)
__global__ __launch_bounds__(256) void make_weight_t(const float* __restrict__ att,
                                                     const float* __restrict__ basis,
                                                     float* __restrict__ WT) {
  int idx = blockIdx.x * 256 + threadIdx.x;  // grid = R*D*D/256 = 4096
  int r = idx >> 16;
  int oi = idx & 65535;
  int o = oi >> 8, i = oi & 255;
  float acc = 0.f;
#pragma unroll
  for (int b = 0; b < BB; ++b) acc += att[r * BB + b] * basis[b * 65536 + i * 256 + o];
  WT[idx] = acc;  // idx = r*65536 + o*256 + i
}

// dst[o,i] = src[i,o] for a 256x256 matrix (root -> rootT)
__global__ __launch_bounds__(256) void transpose_256(const float* __restrict__ src, float* __restrict__ dst) {
  int idx = blockIdx.x * 256 + threadIdx.x;  // grid 256
  int o = idx >> 8, i = idx & 255;
  dst[idx] = src[i * 256 + o];
}

// ---------------- edge message GEMM: 16 same-relation edges per block ----------------
// msg[e,:] = (h[src_e,:] @ W_rel) * ea_e ; atomically accumulated into summ[dst_e,:]
__global__ __launch_bounds__(256) void edge_gemm(const float* __restrict__ h, const int* __restrict__ ei,
                                                 const float* __restrict__ ea, const int* __restrict__ ids,
                                                 const int* __restrict__ off, const float* __restrict__ WT,
                                                 float* __restrict__ summ, float* __restrict__ cnt) {
  __shared__ __align__(16) float lA[16][APAD];
  __shared__ int sSrc[16];
  __shared__ int sDid[16];
  __shared__ float sWe[16];
  __shared__ int sRel;

  const int tid = threadIdx.x;
  const int tstart = blockIdx.x * 16;
  if (tstart >= off[RR]) return;  // uniform over block

  if (tid == 0) {
    int r = 0;
    while (r < RR - 1 && tstart >= off[r + 1]) ++r;
    sRel = r;
  }
  if (tid < 16) {
    int id = ids[tstart + tid];
    int s = 0, d = 0;
    float w = 0.f;
    if (id >= 0) {
      if (id < EE) { s = ei[id]; d = ei[EE + id]; w = ea[id]; }
      else         { s = id - EE; d = s; w = 1.0f; }
      atomicAddF32(&cnt[d], 1.0f);
    }
    sSrc[tid] = s; sDid[tid] = d; sWe[tid] = w;
  }
  __syncthreads();

  // async gather of A = h[src] rows straight into LDS (4x B128 per thread)
  {
    const int r = tid >> 4, seg = tid & 15;
    const float* hp = h + (size_t)sSrc[r] * DD + seg * 4;
#pragma unroll
    for (int p = 0; p < 4; ++p) async_load16(&lA[r][seg * 4 + p * 64], hp + p * 64);
    async_wait0();
  }
  __syncthreads();

  const int wave = tid >> 5, lane = tid & 31;
  const int l16 = lane & 15, hi = lane >> 4;
  const int cb = (wave << 5) + l16;  // wave owns cols [32w,32w+32): two 16-wide tiles
  const float* bt0 = WT + (size_t)sRel * (DD * DD) + (size_t)cb * DD;
  const float* bt1 = bt0 + 16 * DD;

  v8f acc0 = {}; v8f acc1 = {};
  for (int k = 0; k < DD; k += 4) {
    v2f a = *(const v2f*)&lA[l16][k + 2 * hi];        // A: M=lane, K pair by half-wave
    v2f bv0 = *(const v2f*)&bt0[k + 2 * hi];          // B: contiguous K pair in WT
    v2f bv1 = *(const v2f*)&bt1[k + 2 * hi];
    acc0 = __builtin_amdgcn_wmma_f32_16x16x4_f32(false, a, false, bv0, (short)0, acc0, false, false);
    acc1 = __builtin_amdgcn_wmma_f32_16x16x4_f32(false, a, false, bv1, (short)0, acc1, false, false);
  }

  // scale by edge confidence and scatter-add to summ[dst]
#pragma unroll
  for (int v = 0; v < 8; ++v) {
    const int rloc = v + 8 * hi;  // C/D layout: VGPR v holds rows v (lanes 0-15) and v+8 (lanes 16-31)
    const float w = sWe[rloc];
    float* dp = summ + (size_t)sDid[rloc] * DD;
    atomicAddF32(&dp[cb], acc0[v] * w);
    atomicAddF32(&dp[cb + 16], acc1[v] * w);
  }
}

// ---------------- fused: root GEMM + mean-agg + bias + residual + LayerNorm + ReLU ----------------
__global__ __launch_bounds__(256) void finalize_kernel(const float* __restrict__ h, const float* __restrict__ summ,
                                                       const float* __restrict__ cnt, const float* __restrict__ rootT,
                                                       const float* __restrict__ bias, const float* __restrict__ g,
                                                       const float* __restrict__ bb, float* __restrict__ out) {
  __shared__ __align__(16) float lA[16][APAD];
  __shared__ float pre[16][DD + 1];
  __shared__ float ps[16][17], pss[16][17];
  __shared__ float mu_s[16], rs_s[16];

  const int tid = threadIdx.x;
  const int row0 = blockIdx.x * 16;

  {  // async-load 16 input rows (also serves as the residual term)
    const int r = tid >> 4, seg = tid & 15;
    const float* hp = h + (size_t)(row0 + r) * DD + seg * 4;
#pragma unroll
    for (int p = 0; p < 4; ++p) async_load16(&lA[r][seg * 4 + p * 64], hp + p * 64);
    async_wait0();
  }
  __syncthreads();

  const int wave = tid >> 5, lane = tid & 31;
  const int l16 = lane & 15, hi = lane >> 4;
  const int cb = (wave << 5) + l16;
  const float* bt0 = rootT + (size_t)cb * DD;
  const float* bt1 = bt0 + 16 * DD;

  v8f acc0 = {}; v8f acc1 = {};
  for (int k = 0; k < DD; k += 4) {
    v2f a = *(const v2f*)&lA[l16][k + 2 * hi];
    v2f bv0 = *(const v2f*)&bt0[k + 2 * hi];
    v2f bv1 = *(const v2f*)&bt1[k + 2 * hi];
    acc0 = __builtin_amdgcn_wmma_f32_16x16x4_f32(false, a, false, bv0, (short)0, acc0, false, false);
    acc1 = __builtin_amdgcn_wmma_f32_16x16x4_f32(false, a, false, bv1, (short)0, acc1, false, false);
  }

  const float bv0s = bias[cb], bv1s = bias[cb + 16];
#pragma unroll
  for (int v = 0; v < 8; ++v) {
    const int rloc = v + 8 * hi;
    const int row = row0 + rloc;
    const float rc = 1.0f / cnt[row];  // cnt >= 1 via self-loops
    const float p0 = summ[(size_t)row * DD + cb] * rc + acc0[v] + bv0s + lA[rloc][cb];
    const float p1 = summ[(size_t)row * DD + cb + 16] * rc + acc1[v] + bv1s + lA[rloc][cb + 16];
    pre[rloc][cb] = p0;
    pre[rloc][cb + 16] = p1;
  }
  __syncthreads();

  {  // LayerNorm stats: 16 threads per row, 16 elems each
    const int r = tid >> 4, seg = tid & 15;
    float s = 0.f, ss = 0.f;
#pragma unroll
    for (int i = 0; i < 16; ++i) {
      float vv = pre[r][seg * 16 + i];
      s += vv; ss += vv * vv;
    }
    ps[r][seg] = s; pss[r][seg] = ss;
  }
  __syncthreads();
  if (tid < 16) {
    float s1 = 0.f, s2 = 0.f;
#pragma unroll
    for (int j = 0; j < 16; ++j) { s1 += ps[tid][j]; s2 += pss[tid][j]; }
    const float mu = s1 * (1.0f / DD);
    const float var = s2 * (1.0f / DD) - mu * mu;
    mu_s[tid] = mu;
    rs_s[tid] = rsqrtf(var + LN_EPS);
  }
  __syncthreads();
  {
    const int r = tid >> 4, c0 = tid & 15;
    const float mu = mu_s[r], rstd = rs_s[r];
    float* op = out + (size_t)(row0 + r) * DD;
#pragma unroll
    for (int i = 0; i < 16; ++i) {
      const int c = c0 + 16 * i;
      const float vv = (pre[r][c] - mu) * rstd * g[c] + bb[c];
      op[c] = fmaxf(vv, 0.0f);
    }
  }
}

// ---------------- host launcher ----------------
extern "C" void kernel_launch(void* const* d_in, const int* in_sizes, int n_in,
                              void* d_out, int out_size, void* d_ws, size_t ws_size,
                              hipStream_t stream) {
  (void)in_sizes; (void)n_in; (void)out_size; (void)ws_size;

  const float* x      = (const float*)d_in[0];
  const int*   ei     = (const int*)d_in[1];
  const int*   et     = (const int*)d_in[2];
  const float* ea     = (const float*)d_in[3];
  const float* basis1 = (const float*)d_in[4];
  const float* att1   = (const float*)d_in[5];
  const float* root1  = (const float*)d_in[6];
  const float* bias1  = (const float*)d_in[7];
  const float* ln1g   = (const float*)d_in[8];
  const float* ln1b   = (const float*)d_in[9];
  const float* basis2 = (const float*)d_in[10];
  const float* att2   = (const float*)d_in[11];
  const float* root2  = (const float*)d_in[12];
  const float* bias2  = (const float*)d_in[13];
  const float* ln2g   = (const float*)d_in[14];
  const float* ln2b   = (const float*)d_in[15];
  float* out = (float*)d_out;

  // workspace carve-up (~111 MB total)
  char* w = (char*)d_ws;
  auto alloc = [&](size_t bytes) -> char* {
    char* p = w;
    w += (bytes + 255) & ~(size_t)255;
    return p;
  };
  float* WT    = (float*)alloc((size_t)RR * DD * DD * 4);   // 4 MB
  float* rootT = (float*)alloc((size_t)DD * DD * 4);        // 256 KB
  float* summ  = (float*)alloc((size_t)NN * DD * 4);        // 51.2 MB
  float* cnt   = (float*)alloc((size_t)NN * 4);
  float* h1    = (float*)alloc((size_t)NN * DD * 4);        // 51.2 MB
  int*   ids   = (int*)alloc((size_t)IDS_CAP * 4);
  int*   meta  = (int*)alloc(256);                          // cntR[16], off[17], cur[16]
  int* cntR = meta;
  int* off  = meta + 16;
  int* cur  = meta + 33;

  const int n_tiles = IDS_CAP / 16;   // 53141 tiles; unused tail tiles exit early

  // -------- bucket edges by relation (shared by both layers) --------
  fill_i32<<<1, 64, 0, stream>>>(meta, 49, 0);
  fill_i32<<<(IDS_CAP + 255) / 256, 256, 0, stream>>>(ids, IDS_CAP, -1);
  hist_kernel<<<(EE + 255) / 256, 256, 0, stream>>>(et, cntR);
  prefix_kernel<<<1, 1, 0, stream>>>(cntR, off, cur);
  scatter_ids<<<(EE + NN + 255) / 256, 256, 0, stream>>>(et, cur, ids);

  // -------- layer 1 --------
  make_weight_t<<<(RR * DD * DD) / 256, 256, 0, stream>>>(att1, basis1, WT);
  transpose_256<<<(DD * DD) / 256, 256, 0, stream>>>(root1, rootT);
  zero_f32<<<(NN * DD + 255) / 256, 256, 0, stream>>>(summ, NN * DD);
  zero_f32<<<(NN + 255) / 256, 256, 0, stream>>>(cnt, NN);
  edge_gemm<<<n_tiles, 256, 0, stream>>>(x, ei, ea, ids, off, WT, summ, cnt);
  finalize_kernel<<<NN / 16, 256, 0, stream>>>(x, summ, cnt, rootT, bias1, ln1g, ln1b, h1);

  // -------- layer 2 --------
  make_weight_t<<<(RR * DD * DD) / 256, 256, 0, stream>>>(att2, basis2, WT);
  transpose_256<<<(DD * DD) / 256, 256, 0, stream>>>(root2, rootT);
  zero_f32<<<(NN * DD + 255) / 256, 256, 0, stream>>>(summ, NN * DD);
  zero_f32<<<(NN + 255) / 256, 256, 0, stream>>>(cnt, NN);
  edge_gemm<<<n_tiles, 256, 0, stream>>>(h1, ei, ea, ids, off, WT, summ, cnt);
  finalize_kernel<<<NN / 16, 256, 0, stream>>>(h1, summ, cnt, rootT, bias2, ln2g, ln2b, out);
}